// Involution2d_3032246911443
// MI455X (gfx1250) — compile-verified
//
#include <hip/hip_runtime.h>

typedef float v2f __attribute__((ext_vector_type(2)));
typedef float v8f __attribute__((ext_vector_type(8)));
typedef unsigned int v4u __attribute__((ext_vector_type(4)));
typedef int v4i __attribute__((ext_vector_type(4)));
typedef int v8i __attribute__((ext_vector_type(8)));

#define B_    8
#define C_    256
#define G_    4
#define CPG   64
#define H_    64
#define W_    64
#define KK    7
#define PAD_  3
#define HW    (H_ * W_)
#define XROW  76               /* padded LDS x-row stride: 4 zero | 64 data | 8 zero */
#define XBUF  (CPG * XROW)     /* 4864 floats per kh x row-set buffer */
#define BROW  65               /* banded table row stride (65 avoids half-wave bank clash) */
#define NT    24               /* banded K rows: t in [0,24), t=22,23 always zero */
#define BBUF  (NT * BROW)      /* 1560 floats per kh banded-weight buffer */

#if defined(__has_builtin)
#if __has_builtin(__builtin_amdgcn_global_load_async_to_lds_b32) && \
    __has_builtin(__builtin_amdgcn_s_wait_asynccnt)
#define HAVE_ASYNC_LDS 1
#endif
#if __has_builtin(__builtin_amdgcn_tensor_load_to_lds) && \
    __has_builtin(__builtin_amdgcn_s_wait_tensorcnt)
#define HAVE_TDM 1
#endif
#endif
#ifndef HAVE_ASYNC_LDS
#define HAVE_ASYNC_LDS 0
#endif
#ifndef HAVE_TDM
#define HAVE_TDM 0
#endif

#if HAVE_ASYNC_LDS
typedef __attribute__((address_space(1))) int* gas_ptr;
typedef __attribute__((address_space(3))) int* las_ptr;
#define ASYNC_CP(dst, src) __builtin_amdgcn_global_load_async_to_lds_b32( \
        (gas_ptr)(src), (las_ptr)(dst), 0, 0)
#define ASYNC_WAIT() __builtin_amdgcn_s_wait_asynccnt(0)
#else
#define ASYNC_CP(dst, src) (*(dst) = *(src))
#define ASYNC_WAIT() ((void)0)
#endif

// Stage one 64(ch) x 64(col) f32 row-set into LDS with the Tensor Data Mover.
// LDS layout synthesized by TDM padding: pad_interval=5 -> pad after every 64
// DWORDs; pad_amount=11 -> 12 DWORDs inserted => rows at (4 + 76k) floats.
static __device__ __forceinline__ void stage_x(const float* gsrc, float* ldst,
                                               int tid) {
#if HAVE_TDM
    if (tid < 32) {                       // one wave issues the DMA
        const unsigned long long ga = (unsigned long long)(size_t)gsrc;
        const unsigned lda =
            (unsigned)(unsigned long long)(las_ptr)ldst;  // LDS byte offset
        v4u g0;
        g0.x = 1u;                                   // count=1, normal mode
        g0.y = lda;                                  // lds_addr (bytes)
        g0.z = (unsigned)ga;                         // global_addr[31:0]
        g0.w = (unsigned)(ga >> 32) | (2u << 30);    // global_addr[56:32] | type=2
        v8i g1;
        g1[0] = (2 << 16) | (1 << 20) | (5 << 22) | (11 << 25);
        //       data=4B     pad_en     intv=64dw    amount=12dw
        g1[1] = (W_  << 16);   // tensor_dim0 = 64        (bits 79:48)
        g1[2] = (CPG << 16);   // tensor_dim1 = 64 rows   (bits 111:80)
        g1[3] = (W_  << 16);   // tile_dim0   = 64        (bits 127:112)
        g1[4] = CPG;           // tile_dim1   = 64 rows   (bits 143:128)
        g1[5] = HW;            // tensor_dim0_stride = 4096 (bits 207:160)
        g1[6] = 0;
        g1[7] = 0;
        const v4i gz4 = {0, 0, 0, 0};
        const v8i gz8 = {0, 0, 0, 0, 0, 0, 0, 0};
        __builtin_amdgcn_tensor_load_to_lds(g0, g1, gz4, gz4, gz8, 0);
    }
#else
    for (int i = tid; i < CPG * W_; i += 256) {
        const int c = i >> 6, col = i & 63;
        ASYNC_CP(ldst + c * XROW + col, gsrc + (size_t)c * HW + col);
    }
#endif
}

static __device__ __forceinline__ void stage_wait(int tid) {
#if HAVE_TDM
    if (tid < 32) __builtin_amdgcn_s_wait_tensorcnt(0);
#endif
    ASYNC_WAIT();
}

// Block = one (b, g, ho).  Double-buffered, TDM/ASYNC-pipelined per kh:
//   lds_x : 64ch x 64col input row-set (TDM tile, stride-76, pre-zeroed pads)
//   lds_b : 24x64 BANDED weight table  bb[t][wo] = w[kh, t-(wo&15), ho, wo]
// Inner loop is pure ds_load + v_wmma: out[c,wo] += sum_t A[c,t]*B[t,wo].
__global__ __launch_bounds__(256) void invol_wmma_f32(
    const float* __restrict__ x, const float* __restrict__ w,
    const float* __restrict__ bias, float* __restrict__ out)
{
    __shared__ float lds_x[2][XBUF];
    __shared__ float lds_b[2][BBUF];

    const int tid = threadIdx.x;
    const int blk = blockIdx.x;      // (b,g,ho)
    const int ho  = blk & 63;
    const int g   = (blk >> 6) & 3;
    const int b   = blk >> 8;

    const float* xbase = x + (size_t)(b * C_ + g * CPG) * HW;   // group's channel 0
    const float* wsrc  = w + ((size_t)(b * G_ + g) * (KK * KK)) * HW
                           + (size_t)ho * W_;

    // ---- one-time zero init: banded buffers + x halo pads ----
    for (int i = tid; i < 2 * BBUF; i += 256)
        (&lds_b[0][0])[i] = 0.f;
    for (int i = tid; i < 2 * CPG * 12; i += 256) {
        const int buf = i / (CPG * 12);
        const int rem = i - buf * (CPG * 12);
        const int row = rem / 12;
        const int p   = rem - row * 12;
        lds_x[buf][row * XROW + (p < 4 ? p : 60 + p)] = 0.f;  // [0..3] and [64..71]
    }
    __syncthreads();   // zeros strictly before the staging below

    // ---- stage kh = 0 (x row-set if in range, banded weights always) ----
    {
        const int hi = ho - PAD_;
        if (hi >= 0 && hi < H_)
            stage_x(xbase + (size_t)hi * W_, &lds_x[0][4], tid);
        for (int i = tid; i < KK * W_; i += 256) {
            const int kw = i >> 6, wo = i & 63;
            ASYNC_CP(&lds_b[0][(kw + (wo & 15)) * BROW + wo],
                     wsrc + (size_t)kw * HW + wo);
        }
    }
    stage_wait(tid);
    __syncthreads();

    // ---- wave -> tiles: wo-tile (wv&3), ch-tiles 2*(wv>>2)+{0,1} ----
    const int lane = tid & 31;
    const int half = lane >> 4;
    const int l15  = lane & 15;
    const int wv   = tid >> 5;
    const int wo0  = (wv & 3) << 4;
    const int c0a  = (wv >> 2) << 5;

    const int rb0 = (c0a + l15) * XROW;              // lane's channel row in lds_x
    const int rb1 = rb0 + 16 * XROW;
    const int bb0 = (half << 1) * BROW + wo0 + l15;  // lane's base in banded table

    v8f acc0 = {0.f, 0.f, 0.f, 0.f, 0.f, 0.f, 0.f, 0.f};
    v8f acc1 = {0.f, 0.f, 0.f, 0.f, 0.f, 0.f, 0.f, 0.f};

    #pragma unroll
    for (int kh = 0; kh < KK; ++kh) {
        // prefetch next kh's x row-set (TDM) + banded weights (overlaps compute)
        if (kh + 1 < KK) {
            const int khn = kh + 1;
            const int hin = ho + khn - PAD_;
            if (hin >= 0 && hin < H_)
                stage_x(xbase + (size_t)hin * W_, &lds_x[khn & 1][4], tid);
            float* nb = lds_b[khn & 1];
            for (int i = tid; i < KK * W_; i += 256) {
                const int kw = i >> 6, wo = i & 63;
                ASYNC_CP(&nb[(kw + (wo & 15)) * BROW + wo],
                         wsrc + (size_t)(khn * KK + kw) * HW + wo);
            }
        }

        const int hi = ho + kh - PAD_;                   // block-uniform
        if (hi >= 0 && hi < H_) {
            const float* bx = lds_x[kh & 1];
            const float* bw = lds_b[kh & 1];

            #pragma unroll
            for (int q = 0; q < 6; ++q) {
                const int t0 = (q << 2) + (half << 1);   // K indices {t0, t0+1}

                // B: unconditional banded-table pair (offset1 = +BROW)
                const int bi = bb0 + (q << 2) * BROW;
                v2f bm;
                bm.x = bw[bi];
                bm.y = bw[bi + BROW];

                // A: unconditional x pair (halo pads pre-zeroed)
                const int ci = wo0 + t0 + 1;             // = 4 + wo0 + t0 - PAD_
                v2f a0, a1;
                a0.x = bx[rb0 + ci];
                a0.y = bx[rb0 + ci + 1];
                a1.x = bx[rb1 + ci];
                a1.y = bx[rb1 + ci + 1];

                acc0 = __builtin_amdgcn_wmma_f32_16x16x4_f32(
                    false, a0, false, bm, (short)0, acc0, false, false);
                acc1 = __builtin_amdgcn_wmma_f32_16x16x4_f32(
                    false, a1, false, bm, (short)0, acc1, false, false);
            }
        }

        if (kh + 1 < KK) {      // flip buffers: all staged data visible to all waves
            stage_wait(tid);
            __syncthreads();
        }
    }

    // ---- epilogue: D vgpr r -> channel c0a + r + 8*half, column wo0 + l15 ----
    #pragma unroll
    for (int r = 0; r < 8; ++r) {
        const int ca  = c0a + r + (half << 3);
        const int cca = g * CPG + ca;
        const size_t oa = ((size_t)(b * C_ + cca) * H_ + ho) * W_ + wo0 + l15;
        out[oa]                   = acc0[r] + bias[cca];
        out[oa + (size_t)16 * HW] = acc1[r] + bias[cca + 16];
    }
}

extern "C" void kernel_launch(void* const* d_in, const int* in_sizes, int n_in,
                              void* d_out, int out_size, void* d_ws, size_t ws_size,
                              hipStream_t stream) {
    const float* x    = (const float*)d_in[0];  // (8,256,64,64) f32
    const float* wgt  = (const float*)d_in[1];  // (8,4,7,7,64,64) f32
    const float* bias = (const float*)d_in[2];  // (256,) f32
    float* out        = (float*)d_out;          // (8,256,64,64) f32

    // one block per (b, g, ho) = 2048 blocks, 8 waves each
    dim3 grid(B_ * G_ * H_), block(256);
    hipLaunchKernelGGL(invol_wmma_f32, grid, block, 0, stream, x, wgt, bias, out);
}